// PocketEmbedder_85555748536420
// MI455X (gfx1250) — compile-verified
//
#include <hip/hip_runtime.h>
#include <math.h>

// Problem constants (from reference)
#define DD 256     // hidden dim = H*C
#define HH 4       // heads
#define CC 64      // channels per head
#define GG 64      // graphs in batch
#define FF 20      // input features
#define OUTD 256
#define NEG_SLOPE 0.2f

typedef float v2f __attribute__((ext_vector_type(2)));
typedef float v8f __attribute__((ext_vector_type(8)));

// float atomic max via sign-aware int/uint encoding (correct for -inf init)
__device__ __forceinline__ void atomicMaxF(float* addr, float val) {
  if (val >= 0.f) {
    atomicMax((int*)addr, __float_as_int(val));
  } else {
    atomicMin((unsigned int*)addr, __float_as_uint(val));
  }
}

// ---------------------------------------------------------------------------
// Y[M,256] = A[M,256] @ W[256,256] + bias[256], fp32 via V_WMMA_F32_16X16X4_F32
// Block = 256 threads (8 waves) computes 32 rows x 256 cols.
// Wave w owns a 32x32 patch: rows [0,32) x cols [32w,32w+32) as 4 16x16 tiles
// (2 A fragments x 2 B fragments, 4 WMMAs per k-step). A tile staged in LDS
// with float4 loads; B fragments stream from L2-resident weights.
// A frag layout (32-bit A 16x4): lanes 0-15 hold K=k,k+1; lanes 16-31 K=k+2,k+3.
// C/D layout: VGPR v: lanes 0-15 -> M=v, lanes 16-31 -> M=v+8.
// ---------------------------------------------------------------------------
__global__ void gemm256_wmma(const float* __restrict__ A,
                             const float* __restrict__ W,
                             const float* __restrict__ bias,
                             float* __restrict__ Y, int M) {
  __shared__ float As[32 * DD];
  const int tid  = threadIdx.x;
  const int row0 = blockIdx.x * 32;

  // Stage 32x256 A tile: 2048 float4s, 8 per thread, coalesced.
#pragma unroll
  for (int i = 0; i < 8; ++i) {
    int f    = i * 256 + tid;        // float4 index in tile
    int row  = f >> 6;               // 64 float4 per row
    int col4 = f & 63;
    int r = row0 + row;
    float4 v = make_float4(0.f, 0.f, 0.f, 0.f);
    if (r < M) v = *(const float4*)(A + (size_t)r * DD + col4 * 4);
    *(float4*)&As[row * DD + col4 * 4] = v;
  }
  __syncthreads();

  const int wave = tid >> 5;
  const int lane = tid & 31;
  const int m    = lane & 15;          // A row within tile
  const int ksel = (lane >> 4) << 1;   // K sub-offset: 0 (lanes 0-15) or 2 (16-31)
  const int ncol = lane & 15;          // B/C column within tile
  const int c0   = wave * 32;

  v8f acc00 = {}, acc01 = {}, acc10 = {}, acc11 = {};
  for (int k = 0; k < DD; k += 4) {
    v2f a0, a1;
    a0.x = As[m * DD + k + ksel];        a0.y = As[m * DD + k + ksel + 1];
    a1.x = As[(16 + m) * DD + k + ksel]; a1.y = As[(16 + m) * DD + k + ksel + 1];
    const float* wk = W + (size_t)(k + ksel) * DD;
    v2f b0, b1;
    b0.x = wk[c0 + ncol];       b0.y = wk[DD + c0 + ncol];
    b1.x = wk[c0 + 16 + ncol];  b1.y = wk[DD + c0 + 16 + ncol];
    acc00 = __builtin_amdgcn_wmma_f32_16x16x4_f32(false, a0, false, b0,
                                                  (short)0, acc00, false, false);
    acc01 = __builtin_amdgcn_wmma_f32_16x16x4_f32(false, a0, false, b1,
                                                  (short)0, acc01, false, false);
    acc10 = __builtin_amdgcn_wmma_f32_16x16x4_f32(false, a1, false, b0,
                                                  (short)0, acc10, false, false);
    acc11 = __builtin_amdgcn_wmma_f32_16x16x4_f32(false, a1, false, b1,
                                                  (short)0, acc11, false, false);
  }

  const int rsel = (lane >> 4) * 8;
  const int col0 = c0 + ncol, col1 = c0 + 16 + ncol;
  const float bb0 = bias[col0], bb1 = bias[col1];
#pragma unroll
  for (int v = 0; v < 8; ++v) {
    int r = row0 + rsel + v;
    if (r < M) {
      Y[(size_t)r * DD + col0] = acc00[v] + bb0;
      Y[(size_t)r * DD + col1] = acc01[v] + bb1;
    }
    int r2 = r + 16;
    if (r2 < M) {
      Y[(size_t)r2 * DD + col0] = acc10[v] + bb0;
      Y[(size_t)r2 * DD + col1] = acc11[v] + bb1;
    }
  }
}

// ---------------------------------------------------------------------------
// ew[e] = relu( relu(d * w1 + b1) @ w2 + b2 ),  d = ||pos[src]-pos[dst]||
// ---------------------------------------------------------------------------
__global__ void edge_prep(const float* __restrict__ pos, const int* __restrict__ ei,
                          int E, const float* __restrict__ w1, const float* __restrict__ b1,
                          const float* __restrict__ w2, const float* __restrict__ b2,
                          float* __restrict__ ew) {
  int e = blockIdx.x * blockDim.x + threadIdx.x;
  if (e >= E) return;
  int s = ei[e], t = ei[E + e];
  float dx = pos[s * 3 + 0] - pos[t * 3 + 0];
  float dy = pos[s * 3 + 1] - pos[t * 3 + 1];
  float dz = pos[s * 3 + 2] - pos[t * 3 + 2];
  float d = sqrtf(dx * dx + dy * dy + dz * dz);
  float acc = b2[0];
#pragma unroll
  for (int j = 0; j < 32; ++j) {
    float hj = d * w1[j] + b1[j];
    hj = hj > 0.f ? hj : 0.f;
    acc += hj * w2[j];
  }
  ew[e] = acc > 0.f ? acc : 0.f;
}

// h[n,d] = x[n,:] @ w_in[:,d] + b_in[d]   (K=20, negligible cost)
__global__ void lin_in(const float* __restrict__ x, const float* __restrict__ w,
                       const float* __restrict__ b, float* __restrict__ h, int N) {
  int idx = blockIdx.x * blockDim.x + threadIdx.x;
  if (idx >= N * DD) return;
  int n = idx / DD, d = idx % DD;
  float acc = b[d];
  const float* xr = x + n * FF;
#pragma unroll
  for (int k = 0; k < FF; ++k) acc += xr[k] * w[k * DD + d];
  h[idx] = acc;
}

// msg = 0 (float4 granularity), nmax = -inf, nsum = 0
__global__ void layer_init(float4* __restrict__ msg4, float* __restrict__ nmax,
                           float* __restrict__ nsum, int N) {
  int i = blockIdx.x * blockDim.x + threadIdx.x;
  if (i < N * (DD / 4)) msg4[i] = make_float4(0.f, 0.f, 0.f, 0.f);
  if (i < N * HH) { nmax[i] = -__builtin_inff(); nsum[i] = 0.f; }
}

// One wave32 per edge, 8 channels/lane (two float4 loads), head = lane>>3.
// logit[e,h] = att[h,:] . leaky_relu(xl[src] + xr[dst] + ew*we); atomic-max nmax[dst,h]
__global__ void edge_logits(const float* __restrict__ xl, const float* __restrict__ xr,
                            const float* __restrict__ ew, const int* __restrict__ ei, int E,
                            const float* __restrict__ we, const float* __restrict__ att,
                            float* __restrict__ alpha, float* __restrict__ nmax) {
  int tid = threadIdx.x;
  int e = blockIdx.x * 8 + (tid >> 5);
  if (e >= E) return;
  int lane = tid & 31;
  int s = ei[e], t = ei[E + e];
  float w = ew[e];
  int cbase = lane * 8;           // channels [cbase, cbase+8) all in head lane>>3
  int hd = lane >> 3;
  const float4* xls = (const float4*)(xl + (size_t)s * DD + cbase);
  const float4* xrt = (const float4*)(xr + (size_t)t * DD + cbase);
  const float4* wec = (const float4*)(we + cbase);
  const float4* atc = (const float4*)(att + cbase);
  float p = 0.f;
#pragma unroll
  for (int q = 0; q < 2; ++q) {
    float4 a = xls[q], bq = xrt[q], c = wec[q], d = atc[q];
    float g;
    g = a.x + bq.x + w * c.x; g = g > 0.f ? g : NEG_SLOPE * g; p += d.x * g;
    g = a.y + bq.y + w * c.y; g = g > 0.f ? g : NEG_SLOPE * g; p += d.y * g;
    g = a.z + bq.z + w * c.z; g = g > 0.f ? g : NEG_SLOPE * g; p += d.z * g;
    g = a.w + bq.w + w * c.w; g = g > 0.f ? g : NEG_SLOPE * g; p += d.w * g;
  }
  p += __shfl_xor(p, 1, 8);
  p += __shfl_xor(p, 2, 8);
  p += __shfl_xor(p, 4, 8);
  if ((lane & 7) == 0) {
    alpha[(size_t)e * HH + hd] = p;
    atomicMaxF(nmax + (size_t)t * HH + hd, p);
  }
}

// ev = exp(logit - max); nsum[dst,h] += ev; msg[dst,:] += xl[src,:] * ev
__global__ void edge_scatter(const float* __restrict__ xl, const float* __restrict__ alpha,
                             const float* __restrict__ nmax, const int* __restrict__ ei, int E,
                             float* __restrict__ nsum, float* __restrict__ msg) {
  int tid = threadIdx.x;
  int e = blockIdx.x * 8 + (tid >> 5);
  if (e >= E) return;
  int lane = tid & 31;
  int s = ei[e], t = ei[E + e];
  int hd = lane >> 3;
  float ev = expf(alpha[(size_t)e * HH + hd] - nmax[(size_t)t * HH + hd]);
  if ((lane & 7) == 0) atomicAdd(nsum + (size_t)t * HH + hd, ev);
  int cbase = lane * 8;
  const float4* xls = (const float4*)(xl + (size_t)s * DD + cbase);
  float* mt = msg + (size_t)t * DD + cbase;
  float4 u0 = xls[0], u1 = xls[1];
  atomicAdd(mt + 0, u0.x * ev); atomicAdd(mt + 1, u0.y * ev);
  atomicAdd(mt + 2, u0.z * ev); atomicAdd(mt + 3, u0.w * ev);
  atomicAdd(mt + 4, u1.x * ev); atomicAdd(mt + 5, u1.y * ev);
  atomicAdd(mt + 6, u1.z * ev); atomicAdd(mt + 7, u1.w * ev);
}

// h = relu(h + msg/(nsum+eps) + bias), float4 granularity
__global__ void node_finalize(float4* __restrict__ h4, const float4* __restrict__ msg4,
                              const float* __restrict__ nsum, const float4* __restrict__ bias4,
                              int N) {
  int i = blockIdx.x * blockDim.x + threadIdx.x;
  if (i >= N * (DD / 4)) return;
  int n = i / (DD / 4);
  int d4 = i % (DD / 4);          // 4 channels, all within head (d4*4)>>6
  float inv = 1.f / (nsum[n * HH + (d4 >> 4)] + 1e-16f);
  float4 hv = h4[i], mv = msg4[i], bv = bias4[d4];
  float4 o;
  o.x = hv.x + mv.x * inv + bv.x; o.x = o.x > 0.f ? o.x : 0.f;
  o.y = hv.y + mv.y * inv + bv.y; o.y = o.y > 0.f ? o.y : 0.f;
  o.z = hv.z + mv.z * inv + bv.z; o.z = o.z > 0.f ? o.z : 0.f;
  o.w = hv.w + mv.w * inv + bv.w; o.w = o.w > 0.f ? o.w : 0.f;
  h4[i] = o;
}

__global__ void pool_init(float* __restrict__ z, float* __restrict__ gmax,
                          float* __restrict__ gsum) {
  int i = blockIdx.x * blockDim.x + threadIdx.x;
  if (i < GG * 3 * DD) z[i] = 0.f;
  if (i < 3 * GG) { gmax[i] = -__builtin_inff(); gsum[i] = 0.f; }
}

// One wave per node: glog[n] = h[n,:].gate_w + gate_b; atomic-max into gmax[batch[n]]
__global__ void gate_logit(const float* __restrict__ h, const float* __restrict__ gw,
                           const float* __restrict__ gb, const int* __restrict__ batch,
                           float* __restrict__ glog, float* __restrict__ gmax, int N) {
  int tid = threadIdx.x;
  int n = blockIdx.x * 8 + (tid >> 5);
  if (n >= N) return;
  int lane = tid & 31;
  const float4* hn = (const float4*)(h + (size_t)n * DD + lane * 8);
  const float4* gc = (const float4*)(gw + lane * 8);
  float p = 0.f;
#pragma unroll
  for (int q = 0; q < 2; ++q) {
    float4 a = hn[q], b = gc[q];
    p += a.x * b.x + a.y * b.y + a.z * b.z + a.w * b.w;
  }
#pragma unroll
  for (int off = 16; off >= 1; off >>= 1) p += __shfl_xor(p, off, 32);
  if (lane == 0) {
    p += gb[0];
    glog[n] = p;
    atomicMaxF(gmax + batch[n], p);
  }
}

__global__ void gate_exp(float* __restrict__ glog, const float* __restrict__ gmax,
                         float* __restrict__ gsum, const int* __restrict__ batch, int N) {
  int n = blockIdx.x * blockDim.x + threadIdx.x;
  if (n >= N) return;
  int b = batch[n];
  float ev = expf(glog[n] - gmax[b]);
  glog[n] = ev;
  atomicAdd(gsum + b, ev);
}

// z[batch[n], zoff+d] += (ge[n]/(gsum+eps)) * v[n,d]
__global__ void pool_scatter(const float* __restrict__ v, const float* __restrict__ glog,
                             const float* __restrict__ gsum, const int* __restrict__ batch,
                             float* __restrict__ z, int N, int zoff) {
  int i = blockIdx.x * blockDim.x + threadIdx.x;
  if (i >= N * DD) return;
  int n = i / DD, d = i % DD;
  int b = batch[n];
  float g = glog[n] / (gsum[b] + 1e-16f);
  atomicAdd(z + (size_t)b * (3 * DD) + zoff + d, g * v[i]);
}

// Tiny dense layer on G=64 rows: out[g,o] = (relu?) in[g,:] @ W[:,o] + b[o]
__global__ void mlp_dense(const float* __restrict__ in, const float* __restrict__ W,
                          const float* __restrict__ b, float* __restrict__ out,
                          int K, int do_relu) {
  int g = blockIdx.x, o = threadIdx.x;
  const float* inr = in + (size_t)g * K;
  float acc = b[o];
  for (int k = 0; k < K; ++k) acc += inr[k] * W[k * OUTD + o];
  if (do_relu) acc = acc > 0.f ? acc : 0.f;
  out[(size_t)g * OUTD + o] = acc;
}

extern "C" void kernel_launch(void* const* d_in, const int* in_sizes, int n_in,
                              void* d_out, int out_size, void* d_ws, size_t ws_size,
                              hipStream_t stream) {
  const float* x       = (const float*)d_in[0];
  const float* pos     = (const float*)d_in[1];
  const int*   ei      = (const int*)d_in[2];
  const int*   batch   = (const int*)d_in[3];
  const float* mlp_w1  = (const float*)d_in[4];
  const float* mlp_b1  = (const float*)d_in[5];
  const float* mlp_w2  = (const float*)d_in[6];
  const float* mlp_b2  = (const float*)d_in[7];
  const float* w_in    = (const float*)d_in[8];
  const float* b_in    = (const float*)d_in[9];
  const float* gat_wl  = (const float*)d_in[10];
  const float* gat_bl  = (const float*)d_in[11];
  const float* gat_wr  = (const float*)d_in[12];
  const float* gat_br  = (const float*)d_in[13];
  const float* gat_we  = (const float*)d_in[14];
  const float* gat_att = (const float*)d_in[15];
  const float* gat_bias= (const float*)d_in[16];
  const float* gate_w  = (const float*)d_in[17];
  const float* gate_b  = (const float*)d_in[18];
  const float* nn_w    = (const float*)d_in[19];
  const float* nn_b    = (const float*)d_in[20];
  const float* out_w1  = (const float*)d_in[21];
  const float* out_b1  = (const float*)d_in[22];
  const float* out_w2  = (const float*)d_in[23];
  const float* out_b2  = (const float*)d_in[24];

  const int N = in_sizes[0] / FF;
  const int E = in_sizes[2] / 2;

  float* h   = (float*)d_out;            // [N, D]  (returned output #1)
  float* enc = h + (size_t)N * DD;       // [G, OUT] (returned output #2)

  // Workspace carve-out, each region padded to 256B so float4 casts are legal.
  auto aup = [](size_t v) { return (v + 63) & ~(size_t)63; };  // in floats
  float* ws = (float*)d_ws;
  float* ew    = ws;  ws += aup((size_t)E);
  float* xl    = ws;  ws += aup((size_t)N * DD);
  float* xr    = ws;  ws += aup((size_t)N * DD);
  float* msg   = ws;  ws += aup((size_t)N * DD);
  float* alpha = ws;  ws += aup((size_t)E * HH);
  float* nmax  = ws;  ws += aup((size_t)N * HH);
  float* nsum  = ws;  ws += aup((size_t)N * HH);
  float* glog  = ws;  ws += aup((size_t)N);
  float* gmax3 = ws;  ws += aup((size_t)3 * GG);
  float* gsum3 = ws;  ws += aup((size_t)3 * GG);
  float* z     = ws;  ws += aup((size_t)GG * 3 * DD);
  float* t1    = ws;  ws += aup((size_t)GG * OUTD);

  dim3 T(256);
  const int ndBlocks   = (N * DD + 255) / 256;
  const int nd4Blocks  = (N * (DD / 4) + 255) / 256;
  const int gemmBlocks = (N + 31) / 32;
  const int edgeBlocks = (E + 7) / 8;
  const int nodeWBlks  = (N + 7) / 8;

  edge_prep<<<(E + 255) / 256, T, 0, stream>>>(pos, ei, E, mlp_w1, mlp_b1, mlp_w2, mlp_b2, ew);
  lin_in<<<ndBlocks, T, 0, stream>>>(x, w_in, b_in, h, N);
  pool_init<<<(GG * 3 * DD + 255) / 256, T, 0, stream>>>(z, gmax3, gsum3);

  for (int blk = 0; blk < 3; ++blk) {
    for (int j = 0; j < 2; ++j) {
      int i = blk * 2 + j;
      gemm256_wmma<<<gemmBlocks, T, 0, stream>>>(h, gat_wl + (size_t)i * DD * DD,
                                                 gat_bl + i * DD, xl, N);
      gemm256_wmma<<<gemmBlocks, T, 0, stream>>>(h, gat_wr + (size_t)i * DD * DD,
                                                 gat_br + i * DD, xr, N);
      layer_init<<<nd4Blocks, T, 0, stream>>>((float4*)msg, nmax, nsum, N);
      edge_logits<<<edgeBlocks, T, 0, stream>>>(xl, xr, ew, ei, E,
                                                gat_we + (size_t)i * DD,
                                                gat_att + (size_t)i * DD, alpha, nmax);
      edge_scatter<<<edgeBlocks, T, 0, stream>>>(xl, alpha, nmax, ei, E, nsum, msg);
      node_finalize<<<nd4Blocks, T, 0, stream>>>((float4*)h, (const float4*)msg, nsum,
                                                 (const float4*)(gat_bias + i * DD), N);
    }
    gate_logit<<<nodeWBlks, T, 0, stream>>>(h, gate_w + (size_t)blk * DD, gate_b + blk,
                                            batch, glog, gmax3 + blk * GG, N);
    gate_exp<<<(N + 255) / 256, T, 0, stream>>>(glog, gmax3 + blk * GG,
                                                gsum3 + blk * GG, batch, N);
    gemm256_wmma<<<gemmBlocks, T, 0, stream>>>(h, nn_w + (size_t)blk * DD * DD,
                                               nn_b + blk * DD, xl, N);
    pool_scatter<<<ndBlocks, T, 0, stream>>>(xl, glog, gsum3 + blk * GG, batch, z, N, blk * DD);
  }

  mlp_dense<<<GG, T, 0, stream>>>(z, out_w1, out_b1, t1, 3 * DD, 1);
  mlp_dense<<<GG, T, 0, stream>>>(t1, out_w2, out_b2, enc, OUTD, 0);
}